// NodeGCN_5798205849977
// MI455X (gfx1250) — compile-verified
//
#include <hip/hip_runtime.h>
#include <math.h>

#define IN_F  128
#define HID_F 128
#define OUT_F 64

typedef __attribute__((ext_vector_type(2))) float v2f;
typedef __attribute__((ext_vector_type(8))) float v8f;

// ---------------- degree / normalization ----------------

__global__ void k_init_deg(float* __restrict__ deg, int n) {
    int i = blockIdx.x * blockDim.x + threadIdx.x;
    if (i < n) deg[i] = 1.0f;   // self-loop contributes 1 to every node's degree
}

__global__ void k_count_deg(const int* __restrict__ dst, float* __restrict__ deg, int E) {
    int i = blockIdx.x * blockDim.x + threadIdx.x;
    if (i < E) atomicAdd(&deg[dst[i]], 1.0f);
}

__global__ void k_rsqrt(float* __restrict__ deg, int n) {
    int i = blockIdx.x * blockDim.x + threadIdx.x;
    if (i < n) {
        float d = deg[i];
        deg[i] = (d > 0.0f) ? rsqrtf(d) : 0.0f;   // deg >= 1 always here
    }
}

// ---------------- fp32 WMMA GEMM: C[M x NC] = A[M x K] @ B[K x NC] ----------
// One wave per 16x16 output tile, V_WMMA_F32_16X16X4_F32, K fully unrolled.
// K and NC are compile-time so every global_load gets an immediate offset from
// a single hoisted base — minimal per-iteration address VALU/SALU.

template <int K, int NC>
__global__ void __launch_bounds__(32)
k_gemm_wmma_f32(const float* __restrict__ A, const float* __restrict__ B,
                float* __restrict__ C, int M) {
    const int lane = threadIdx.x;          // 0..31
    const int half = lane >> 4;            // 0: K+0/1 , 1: K+2/3
    const int l    = lane & 15;

    const int row0 = blockIdx.x * 16;
    const int col0 = blockIdx.y * 16;

    int arow = row0 + l;
    if (arow >= M) arow = M - 1;           // clamp loads; stores are guarded
    const float* __restrict__ Ap = A + (size_t)arow * K + half * 2;   // + k gives A frag
    const float* __restrict__ Bp = B + (size_t)(half * 2) * NC + col0 + l;

    v8f acc = {};
    #pragma unroll
    for (int k = 0; k < K; k += 4) {
        // A fragment: 16x4, lane row = l, two consecutive K values (aligned float2)
        v2f a;
        {
            const float2 av = *(const float2*)(Ap + k);
            a.x = av.x; a.y = av.y;
        }
        // B fragment: 4x16, lane col = l, same two K rows (immediate offsets)
        v2f b;
        b.x = Bp[(size_t)k * NC];
        b.y = Bp[(size_t)k * NC + NC];

        acc = __builtin_amdgcn_wmma_f32_16x16x4_f32(
            /*neg_a=*/false, a, /*neg_b=*/false, b,
            /*c_mod=*/(short)0, acc, /*reuse_a=*/false, /*reuse_b=*/false);
    }

    // D layout: VGPR v -> M = row0 + v + 8*half, N = col0 + (lane&15)
    float* __restrict__ Cp = C + (size_t)(row0 + half * 8) * NC + col0 + l;
    #pragma unroll
    for (int v = 0; v < 8; ++v) {
        if (row0 + half * 8 + v < M) Cp[(size_t)v * NC] = acc[v];
    }
}

// ---------------- bias + self-loop message (norm = dinv[i]^2) ---------------

template <int C>
__global__ void k_bias_selfloop(const float* __restrict__ h, const float* __restrict__ bias,
                                const float* __restrict__ dinv, float* __restrict__ out,
                                int n) {
    long long idx = (long long)blockIdx.x * blockDim.x + threadIdx.x;
    if (idx >= (long long)n * C) return;
    const int i = (int)(idx / C);
    const int c = (int)(idx % C);
    const float di = dinv[i];
    out[idx] = bias[c] + h[idx] * di * di;
}

// ---------------- edge gather/scale/scatter-add -----------------------------
// One wave per edge; lane covers C/32 contiguous features. Feature matrix
// (51 MB for C=128) fits in MI455X's 192 MB L2, so both the random gather and
// the float atomics should stay L2-resident.

template <int C>
__global__ void k_scatter(const float* __restrict__ h, const int* __restrict__ src,
                          const int* __restrict__ dst, const float* __restrict__ dinv,
                          float* __restrict__ out, int E) {
    const int e    = (int)((blockIdx.x * blockDim.x + threadIdx.x) >> 5);
    const int lane = threadIdx.x & 31;
    if (e >= E) return;
    const int s = src[e];
    const int d = dst[e];
    const float nrm = dinv[s] * dinv[d];
    constexpr int V = C / 32;
    const float* hp = h   + (size_t)s * C + lane * V;
    float*       op = out + (size_t)d * C + lane * V;
    #pragma unroll
    for (int j = 0; j < V; ++j) atomicAdd(op + j, hp[j] * nrm);
}

// ---------------- per-row softmax over 128 features (one wave per node) -----

__global__ void k_softmax128(float* __restrict__ x, int n) {
    const int wavesPerBlock = blockDim.x >> 5;
    const int node = blockIdx.x * wavesPerBlock + (threadIdx.x >> 5);
    const int lane = threadIdx.x & 31;
    if (node >= n) return;
    float* row = x + (size_t)node * 128;

    float4 v = *(const float4*)(row + lane * 4);
    float m = fmaxf(fmaxf(v.x, v.y), fmaxf(v.z, v.w));
    #pragma unroll
    for (int off = 16; off > 0; off >>= 1) m = fmaxf(m, __shfl_xor(m, off, 32));

    v.x = expf(v.x - m); v.y = expf(v.y - m);
    v.z = expf(v.z - m); v.w = expf(v.w - m);
    float s = v.x + v.y + v.z + v.w;
    #pragma unroll
    for (int off = 16; off > 0; off >>= 1) s += __shfl_xor(s, off, 32);

    const float inv = 1.0f / s;
    v.x *= inv; v.y *= inv; v.z *= inv; v.w *= inv;
    *(float4*)(row + lane * 4) = v;
}

// ---------------- launch ----------------------------------------------------

extern "C" void kernel_launch(void* const* d_in, const int* in_sizes, int n_in,
                              void* d_out, int out_size, void* d_ws, size_t ws_size,
                              hipStream_t stream) {
    const float* x  = (const float*)d_in[0];
    const float* W1 = (const float*)d_in[1];
    const float* b1 = (const float*)d_in[2];
    const float* W2 = (const float*)d_in[3];
    const float* b2 = (const float*)d_in[4];
    const int*   ei = (const int*)  d_in[5];

    const int n = in_sizes[0] / IN_F;
    const int E = in_sizes[5] / 2;
    const int* src  = ei;
    const int* dstv = ei + E;

    float* ws   = (float*)d_ws;
    float* dinv = ws;                              // n floats (deg -> dinv in place)
    float* h1   = dinv + n;                        // n*128
    float* agg1 = h1 + (size_t)n * HID_F;          // n*128
    float* h2   = h1;                              // alias: h1 is dead after scatter1
    float* out  = (float*)d_out;                   // n*64

    const int T = 256;

    // 1) degree + symmetric normalization
    k_init_deg <<<(n + T - 1) / T, T, 0, stream>>>(dinv, n);
    k_count_deg<<<(E + T - 1) / T, T, 0, stream>>>(dstv, dinv, E);
    k_rsqrt    <<<(n + T - 1) / T, T, 0, stream>>>(dinv, n);

    // 2) h1 = x @ W1   (WMMA fp32)
    {
        dim3 grid((n + 15) / 16, HID_F / 16);
        k_gemm_wmma_f32<IN_F, HID_F><<<grid, 32, 0, stream>>>(x, W1, h1, n);
    }

    // 3) agg1 = b1 + selfloop(h1);   scatter edges into agg1
    {
        long long tot = (long long)n * HID_F;
        k_bias_selfloop<HID_F><<<(unsigned)((tot + T - 1) / T), T, 0, stream>>>(h1, b1, dinv, agg1, n);
        long long thr = (long long)E * 32;
        k_scatter<HID_F><<<(unsigned)((thr + T - 1) / T), T, 0, stream>>>(h1, src, dstv, dinv, agg1, E);
    }

    // 4) row softmax in place on agg1
    k_softmax128<<<(n + (T / 32) - 1) / (T / 32), T, 0, stream>>>(agg1, n);

    // 5) h2 = softmax(agg1) @ W2   (WMMA fp32)
    {
        dim3 grid((n + 15) / 16, OUT_F / 16);
        k_gemm_wmma_f32<HID_F, OUT_F><<<grid, 32, 0, stream>>>(agg1, W2, h2, n);
    }

    // 6) out = b2 + selfloop(h2);   scatter edges into out
    {
        long long tot = (long long)n * OUT_F;
        k_bias_selfloop<OUT_F><<<(unsigned)((tot + T - 1) / T), T, 0, stream>>>(h2, b2, dinv, out, n);
        long long thr = (long long)E * 32;
        k_scatter<OUT_F><<<(unsigned)((thr + T - 1) / T), T, 0, stream>>>(h2, src, dstv, dinv, out, E);
    }
}